// OutMultiChoices_action_19292993093894
// MI455X (gfx1250) — compile-verified
//
#include <hip/hip_runtime.h>
#include <math.h>

#define E     512
#define TWOE  1024
#define SIXE  3072
#define BATCH 1280
#define TQ    64
#define TA    32
#define TV    64
#define BN_EPS 1e-5f

typedef float v2f __attribute__((ext_vector_type(2)));
typedef float v4f __attribute__((ext_vector_type(4)));
typedef float v8f __attribute__((ext_vector_type(8)));

// ---------------------------------------------------------------------------
// Phase 1: (masked) mean over T of x[T, B, 1024] -> out[b*outStride + d]
// One thread per float4 of the [B,1024] output; loop runs only to len[b]
// (uniform per wave: 256 threads share one b => 8 full waves per b).
// Nontemporal B128 loads: data is streamed exactly once, keep L2 for weights.
// ---------------------------------------------------------------------------
__global__ void mean_over_t(const float* __restrict__ x,
                            const int* __restrict__ lens,
                            float* __restrict__ out,
                            int T, int outStride) {
  const int NV = TWOE / 4;
  int tid = blockIdx.x * blockDim.x + threadIdx.x;
  int b = tid / NV;
  int v = tid % NV;
  if (b >= BATCH) return;

  int len = T;
  if (lens) {
    len = lens[b];
    if (len < 1) len = 1;
    if (len > T) len = T;
  }

  const v4f* p = (const v4f*)(x + (size_t)b * TWOE) + v;
  const size_t stride = (size_t)BATCH * TWOE / 4;   // in float4 units

  v4f acc = {0.f, 0.f, 0.f, 0.f};
  for (int t = 0; t < len; ++t) {
    v4f d = __builtin_nontemporal_load(&p[(size_t)t * stride]);
    acc += d;
  }
  float inv = 1.0f / (float)len;
  acc *= inv;
  *((v4f*)(out + (size_t)b * outStride) + v) = acc;
}

// ---------------------------------------------------------------------------
// Phase 2: fp32 WMMA GEMM  D = act(A[MxK] @ W[KxN] + bias), all row-major.
// One wave per 16x16 tile, K stepped by 4 via V_WMMA_F32_16X16X4_F32.
// A-frag (16x4):  lane l<16 -> row M=l, K=k0+0..1 ; lane l>=16 -> K=k0+2..3
// B-frag (4x16):  VGPR i, lane l(<16) -> B[k0+i][n0+l]; lanes>=16 -> K+2
// D (16x16 f32):  VGPR i, lanes 0-15 -> row i, lanes 16-31 -> row i+8
// Tile indices are wave-uniform so EXEC stays all-ones around the WMMA.
// ---------------------------------------------------------------------------
__global__ void gemm_wmma_f32(const float* __restrict__ A, int lda,
                              const float* __restrict__ W, int ldw,
                              const float* __restrict__ bias,
                              float* __restrict__ Dst, int ldd,
                              int M, int N, int K, int elu) {
  int gid  = blockIdx.x * blockDim.x + threadIdx.x;
  int wave = gid >> 5;
  int lane = gid & 31;
  int tilesN = N >> 4;
  int tm = wave / tilesN;
  int tn = wave % tilesN;
  if ((tm << 4) >= M) return;                 // whole-wave exit only

  int m0 = tm << 4, n0 = tn << 4;
  int half = lane >> 4;                       // 0 or 1
  int l    = lane & 15;

  float bv = bias ? bias[n0 + l] : 0.0f;
  v8f c = {bv, bv, bv, bv, bv, bv, bv, bv};

  const float* arow = A + (size_t)(m0 + l) * lda;
  for (int k0 = 0; k0 < K; k0 += 4) {
    int kk = k0 + half * 2;                   // 8B-aligned (kk even)
    v2f a = *(const v2f*)(arow + kk);
    const float* wp = W + (size_t)kk * ldw + n0 + l;
    v2f b;
    b.x = wp[0];
    b.y = wp[ldw];
    c = __builtin_amdgcn_wmma_f32_16x16x4_f32(
            /*neg_a=*/false, a, /*neg_b=*/false, b,
            /*c_mod=*/(short)0, c, /*reuse_a=*/false, /*reuse_b=*/false);
  }

  int rbase = m0 + half * 8;
  float* dcol = Dst + (size_t)rbase * ldd + n0 + l;
#pragma unroll
  for (int i = 0; i < 8; ++i) {
    float v = c[i];
    if (elu) v = (v > 0.0f) ? v : expm1f(v);  // ELU(alpha=1)
    dcol[(size_t)i * ldd] = v;
  }
}

// ---------------------------------------------------------------------------
// Phase 3a: per-feature batch stats over h[B,512] (L2-resident, 2.6 MB),
// folded with gamma/beta/Wc2 into dot coefficients:
//   c1[d]    = gamma[d] * rsqrt(var+eps) * Wc2[d]
//   c0vec[d] = (beta[d] - mu * gamma[d] * rsqrt(var+eps)) * Wc2[d]
// ---------------------------------------------------------------------------
__global__ void bn_coeffs(const float* __restrict__ h,
                          const float* __restrict__ gamma,
                          const float* __restrict__ beta,
                          const float* __restrict__ Wc2,
                          float* __restrict__ c1,
                          float* __restrict__ c0vec) {
  int d = blockIdx.x;
  int t = threadIdx.x;
  float s = 0.f, s2 = 0.f;
  for (int b = t; b < BATCH; b += blockDim.x) {
    float v = h[(size_t)b * E + d];
    s += v; s2 += v * v;
  }
  __shared__ float ss[256], sq[256];
  ss[t] = s; sq[t] = s2;
  __syncthreads();
  for (int off = blockDim.x >> 1; off > 0; off >>= 1) {
    if (t < off) { ss[t] += ss[t + off]; sq[t] += sq[t + off]; }
    __syncthreads();
  }
  if (t == 0) {
    float mu  = ss[0] / (float)BATCH;
    float var = sq[0] / (float)BATCH - mu * mu;   // biased, matches jnp.var
    float inv = rsqrtf(var + BN_EPS);
    float g = gamma[d] * inv;
    float w = Wc2[d];
    c1[d]    = g * w;
    c0vec[d] = (beta[d] - mu * g) * w;
  }
}

// ---------------------------------------------------------------------------
// Phase 3b: out[b] = dot(h[b,:], c1) + sum(c0vec) + bc2. One wave32 per row.
// ---------------------------------------------------------------------------
__global__ void final_dot(const float* __restrict__ h,
                          const float* __restrict__ c1,
                          const float* __restrict__ c0vec,
                          const float* __restrict__ bc2,
                          float* __restrict__ out) {
  int gid  = blockIdx.x * blockDim.x + threadIdx.x;
  int wave = gid >> 5;
  int lane = gid & 31;
  if (wave >= BATCH) return;
  const float* hr = h + (size_t)wave * E;
  float acc = 0.f;
  for (int d = lane; d < E; d += 32)
    acc += hr[d] * c1[d] + c0vec[d];
  for (int off = 16; off > 0; off >>= 1)
    acc += __shfl_xor(acc, off, 32);
  if (lane == 0) out[wave] = acc + bc2[0];
}

// ---------------------------------------------------------------------------
extern "C" void kernel_launch(void* const* d_in, const int* in_sizes, int n_in,
                              void* d_out, int out_size, void* d_ws, size_t ws_size,
                              hipStream_t stream) {
  const float* qe    = (const float*)d_in[0];   // [64,1280,1024]
  const float* vqe   = (const float*)d_in[1];   // [64,1280,1024]
  const float* ae    = (const float*)d_in[2];   // [32,1280,1024]
  const float* vae   = (const float*)d_in[3];   // [64,1280,1024]
  const int*   qlen  = (const int*)d_in[4];     // [1280]
  const int*   alen  = (const int*)d_in[5];     // [1280]
  const float* Wq    = (const float*)d_in[6];   // [1024,512]
  const float* bq    = (const float*)d_in[7];
  const float* Wa    = (const float*)d_in[8];   // [1024,512]
  const float* ba    = (const float*)d_in[9];
  const float* Wc1   = (const float*)d_in[10];  // [3072,512]
  const float* bc1   = (const float*)d_in[11];
  const float* gamma = (const float*)d_in[12];
  const float* beta  = (const float*)d_in[13];
  const float* Wc2   = (const float*)d_in[14];  // [512,1]
  const float* bc2   = (const float*)d_in[15];  // [1]
  float* out = (float*)d_out;

  // Scratch layout (floats): mq[B,1024] | ma[B,1024] | cat[B,3072] | h[B,512] | c1[512] | c0vec[512]
  float* ws  = (float*)d_ws;
  float* mq  = ws;
  float* ma  = mq  + (size_t)BATCH * TWOE;
  float* cat = ma  + (size_t)BATCH * TWOE;
  float* h   = cat + (size_t)BATCH * SIXE;
  float* c1  = h   + (size_t)BATCH * E;
  float* c0v = c1  + E;

  dim3 blk(256);

  // Phase 1: four streaming mean reductions.
  // cat columns: [ q 0:512 | vq 512:1536 | a 1536:2048 | va 2048:3072 ]
  int meanBlocks = (BATCH * (TWOE / 4) + 255) / 256;
  mean_over_t<<<meanBlocks, blk, 0, stream>>>(qe,  qlen,    mq,           TQ, TWOE);
  mean_over_t<<<meanBlocks, blk, 0, stream>>>(vqe, nullptr, cat + E,      TV, SIXE);
  mean_over_t<<<meanBlocks, blk, 0, stream>>>(ae,  alen,    ma,           TA, TWOE);
  mean_over_t<<<meanBlocks, blk, 0, stream>>>(vae, nullptr, cat + 4 * E,  TV, SIXE);

  // Phase 2: WMMA GEMMs (M=1280, N=512): waves = 80*32 = 2560.
  int gemmBlocks = ((BATCH / 16) * (E / 16) * 32 + 255) / 256;
  gemm_wmma_f32<<<gemmBlocks, blk, 0, stream>>>(mq,  TWOE, Wq,  E, bq,  cat,         SIXE, BATCH, E, TWOE, 0);
  gemm_wmma_f32<<<gemmBlocks, blk, 0, stream>>>(ma,  TWOE, Wa,  E, ba,  cat + 3 * E, SIXE, BATCH, E, TWOE, 0);
  gemm_wmma_f32<<<gemmBlocks, blk, 0, stream>>>(cat, SIXE, Wc1, E, bc1, h,           E,    BATCH, E, SIXE, 1);

  // Phase 3: BatchNorm folded into dot coefficients, then one wave per row.
  bn_coeffs<<<E, blk, 0, stream>>>(h, gamma, beta, Wc2, c1, c0v);
  final_dot<<<(BATCH * 32 + 255) / 256, blk, 0, stream>>>(h, c1, c0v, bc2, out);
}